// FactorizedVectorQuantize_274877907453
// MI455X (gfx1250) — compile-verified
//
#include <hip/hip_runtime.h>
#include <hip/hip_bf16.h>

typedef __attribute__((ext_vector_type(2))) float v2f;
typedef __attribute__((ext_vector_type(8))) float v8f;

#define B_ 8
#define D_IN_ 1024
#define T_ 2048
#define CB_SIZE_ 8192
#define CB_DIM_ 8
#define NPOS_ (B_*T_)            // 16384
#define EPS_ 1e-12f

// ---- d_out layout (concatenated outputs, float32) ----
#define OUT_OFF   ((size_t)0)
#define IDX_OFF   ((size_t)B_*D_IN_*T_)              // 16777216
#define DIST_OFF  (IDX_OFF + (size_t)B_*T_)          // 16793600
#define VQ_OFF    (DIST_OFF + (size_t)NPOS_*CB_SIZE_)
#define PPL_OFF   (VQ_OFF + 1)
#define ACT_OFF   (VQ_OFF + 2)

// ---- workspace layout (in floats) ----
enum : size_t {
  WS_WIN  = 0,                    // w_in  [8,1024]
  WS_WOUT = WS_WIN  + 8192,       // w_out [1024,8]
  WS_CBN  = WS_WOUT + 8192,       // cb_n  [8192,8]
  WS_SSB  = WS_CBN  + 65536,      // ssB   [8192]
  WS_ZE   = WS_SSB  + 8192,       // z_e   [16384,8]
  WS_ENC  = WS_ZE   + 131072,     // enc_n [16384,8]
  WS_SSA  = WS_ENC  + 131072,     // ssA   [16384]
  WS_IDX  = WS_SSA  + 16384,      // idx   [16384] int
  WS_CNT  = WS_IDX  + 16384,      // counts[8192] uint
  WS_PART = WS_CNT  + 8192        // partials[64]
};

// ---------------- weight norm: v_in (norm over 1024 per channel) ----------------
__global__ void k_wn_in(const float* __restrict__ v_in, const float* __restrict__ g_in,
                        float* __restrict__ w_in) {
  int c = blockIdx.x;
  __shared__ float red[256];
  float s = 0.f;
  for (int d = threadIdx.x; d < D_IN_; d += 256) {
    float v = v_in[c*D_IN_ + d];
    s += v*v;
  }
  red[threadIdx.x] = s;
  __syncthreads();
  for (int o = 128; o > 0; o >>= 1) {
    if (threadIdx.x < o) red[threadIdx.x] += red[threadIdx.x + o];
    __syncthreads();
  }
  float scale = g_in[c] / fmaxf(sqrtf(red[0]), EPS_);
  for (int d = threadIdx.x; d < D_IN_; d += 256)
    w_in[c*D_IN_ + d] = v_in[c*D_IN_ + d] * scale;
}

// ---------------- row-wise norms: w_out rows + normalized codebook ----------------
__global__ void k_wn_rows(const float* __restrict__ v_out, const float* __restrict__ g_out,
                          const float* __restrict__ codebook,
                          float* __restrict__ w_out, float* __restrict__ cb_n,
                          float* __restrict__ ssB) {
  int gid = blockIdx.x*256 + threadIdx.x;          // 0..9215
  if (gid < D_IN_) {
    float r[8]; float s = 0.f;
    #pragma unroll
    for (int c = 0; c < 8; ++c) { r[c] = v_out[gid*8 + c]; s += r[c]*r[c]; }
    float sc = g_out[gid] / fmaxf(sqrtf(s), EPS_);
    #pragma unroll
    for (int c = 0; c < 8; ++c) w_out[gid*8 + c] = r[c]*sc;
  } else {
    int j = gid - D_IN_;
    float r[8]; float s = 0.f;
    #pragma unroll
    for (int c = 0; c < 8; ++c) { r[c] = codebook[j*8 + c]; s += r[c]*r[c]; }
    float inv = 1.f / fmaxf(sqrtf(s), EPS_);
    float ss = 0.f;
    #pragma unroll
    for (int c = 0; c < 8; ++c) { float x = r[c]*inv; cb_n[j*8+c] = x; ss += x*x; }
    ssB[j] = ss;
  }
}

// ---------------- z_e = w_in . z + b_in ; enc_n ; ssA ----------------
// block: 256 threads = 64 positions x 4 d-quarters (wave-contiguous t for coalescing)
__global__ void k_zproj(const float* __restrict__ z, const float* __restrict__ w_in,
                        const float* __restrict__ b_in,
                        float* __restrict__ z_e, float* __restrict__ enc_n,
                        float* __restrict__ ssA) {
  __shared__ __align__(16) float wl[D_IN_*8];      // [d][c], 32 KB
  __shared__ float pacc[256][8];
  for (int i = threadIdx.x; i < D_IN_*8; i += 256) {
    int d = i >> 3, c = i & 7;
    wl[i] = w_in[c*D_IN_ + d];
  }
  __syncthreads();
  const int q  = threadIdx.x >> 6;                 // 0..3  (d quarter)
  const int tl = threadIdx.x & 63;                 // 0..63 (position in block)
  const int p = blockIdx.x*64 + tl;
  const int b = p >> 11;
  const int t = p & 2047;
  const float* zp = z + (size_t)b*D_IN_*T_ + t;
  float acc[8];
  #pragma unroll
  for (int c = 0; c < 8; ++c) acc[c] = 0.f;
  const int d0 = q*256;
  for (int dd = 0; dd < 256; ++dd) {
    int d = d0 + dd;
    float zv = __builtin_nontemporal_load(&zp[(size_t)d*T_]);   // z read once: NT
    const float* w = &wl[d*8];
    #pragma unroll
    for (int c = 0; c < 8; ++c) acc[c] += w[c]*zv;
  }
  #pragma unroll
  for (int c = 0; c < 8; ++c) pacc[threadIdx.x][c] = acc[c];
  __syncthreads();
  if (q == 0) {
    float ze[8]; float ss = 0.f;
    #pragma unroll
    for (int c = 0; c < 8; ++c) {
      float s = pacc[tl][c] + pacc[tl+64][c] + pacc[tl+128][c] + pacc[tl+192][c];
      s += b_in[c];
      ze[c] = s; ss += s*s;
    }
    float inv = 1.f / fmaxf(sqrtf(ss), EPS_);
    float sa = 0.f;
    #pragma unroll
    for (int c = 0; c < 8; ++c) {
      z_e[p*8+c] = ze[c];
      float e = ze[c]*inv;
      enc_n[p*8+c] = e;
      sa += e*e;
    }
    ssA[p] = sa;
  }
}

// ---------------- dist (WMMA f32 16x16x4, K=8 via 2 chained) + argmin + counts ----------------
// grid: 1024 position-tiles of 16; block: 8 waves, each wave covers 1024 codebook cols
__global__ void k_dist(const float* __restrict__ enc_n, const float* __restrict__ cb_n,
                       const float* __restrict__ ssA, const float* __restrict__ ssB,
                       float* __restrict__ dist, float* __restrict__ idx_out_f,
                       int* __restrict__ idx_i, unsigned int* __restrict__ counts) {
  __shared__ float smin[8][16];
  __shared__ int   sidx[8][16];
  const int tid  = threadIdx.x;
  const int wave = tid >> 5;
  const int lane = tid & 31;
  const int half = lane >> 4;          // 0: rows 0-7 / K0-1,  1: rows 8-15 / K2-3
  const int nrow = lane & 15;          // row of A / col of B
  const int koff = half*2;
  const int p0 = blockIdx.x*16;

  // A tile: enc_n rows p0..p0+15, K=8 (split into two K=4 chunks)
  const float* arow = enc_n + (size_t)(p0 + nrow)*8;
  v2f a1, a2;
  a1.x = arow[koff];   a1.y = arow[koff+1];
  a2.x = arow[4+koff]; a2.y = arow[5+koff];

  float sa[8];
  #pragma unroll
  for (int r = 0; r < 8; ++r) sa[r] = ssA[p0 + half*8 + r];

  float rmin[8]; int ridx[8];
  #pragma unroll
  for (int r = 0; r < 8; ++r) { rmin[r] = 3.4e38f; ridx[r] = 0x7fffffff; }

  const int jbase = wave*1024;
  #pragma unroll 2
  for (int jt = 0; jt < 64; ++jt) {
    const int j0 = jbase + jt*16;
    const float* brow = cb_n + (size_t)(j0 + nrow)*8;
    v2f b1, b2;
    b1.x = brow[koff];   b1.y = brow[koff+1];
    b2.x = brow[4+koff]; b2.y = brow[5+koff];
    const float sb = ssB[j0 + nrow];
    v8f c = {};
    c = __builtin_amdgcn_wmma_f32_16x16x4_f32(false, a1, false, b1, (short)0, c, false, false);
    c = __builtin_amdgcn_wmma_f32_16x16x4_f32(false, a2, false, b2, (short)0, c, false, false);
    float* drow = dist + (size_t)(p0 + half*8)*CB_SIZE_ + j0 + nrow;
    const int ji = j0 + nrow;
    #pragma unroll
    for (int r = 0; r < 8; ++r) {
      float dv = sa[r] + sb - 2.0f*c[r];
      __builtin_nontemporal_store(dv, &drow[(size_t)r*CB_SIZE_]);  // dist: write-once stream
      if (dv < rmin[r]) { rmin[r] = dv; ridx[r] = ji; }
    }
  }
  // reduce argmin across the 16 lanes of each half (xor stays within half)
  #pragma unroll
  for (int off = 1; off < 16; off <<= 1) {
    #pragma unroll
    for (int r = 0; r < 8; ++r) {
      float ov = __shfl_xor(rmin[r], off, 32);
      int   oi = __shfl_xor(ridx[r], off, 32);
      if (ov < rmin[r] || (ov == rmin[r] && oi < ridx[r])) { rmin[r] = ov; ridx[r] = oi; }
    }
  }
  if (nrow == 0) {
    #pragma unroll
    for (int r = 0; r < 8; ++r) {
      smin[wave][half*8 + r] = rmin[r];
      sidx[wave][half*8 + r] = ridx[r];
    }
  }
  __syncthreads();
  if (tid < 16) {
    float bv = smin[0][tid]; int bi = sidx[0][tid];
    #pragma unroll
    for (int w = 1; w < 8; ++w) {
      float v = smin[w][tid]; int i2 = sidx[w][tid];
      if (v < bv || (v == bv && i2 < bi)) { bv = v; bi = i2; }
    }
    const int p = p0 + tid;
    idx_i[p] = bi;
    idx_out_f[p] = (float)bi;
    atomicAdd(&counts[bi], 1u);
  }
}

// ---------------- vq loss partial sums ----------------
__global__ void k_vqpart(const float* __restrict__ z_e, const float* __restrict__ codebook,
                         const int* __restrict__ idx_i, float* __restrict__ partials) {
  __shared__ float red[256];
  int p = blockIdx.x*256 + threadIdx.x;
  int j = idx_i[p];
  float e = 0.f;
  #pragma unroll
  for (int c = 0; c < 8; ++c) {
    float d = z_e[p*8+c] - codebook[j*8+c];
    e += d*d;
  }
  red[threadIdx.x] = e;
  __syncthreads();
  for (int o = 128; o > 0; o >>= 1) {
    if (threadIdx.x < o) red[threadIdx.x] += red[threadIdx.x+o];
    __syncthreads();
  }
  if (threadIdx.x == 0) partials[blockIdx.x] = red[0];
}

// ---------------- finalize scalars ----------------
__global__ void k_finalize(const unsigned int* __restrict__ counts,
                           const float* __restrict__ partials,
                           float* __restrict__ vq_out, float* __restrict__ ppl_out,
                           float* __restrict__ act_out) {
  __shared__ float rent[256], ract[256], rvq[256];
  float ent = 0.f, act = 0.f, vq = 0.f;
  for (int j = threadIdx.x; j < CB_SIZE_; j += 256) {
    float cnt = (float)counts[j];
    float avg = cnt / (float)NPOS_;
    ent += avg * logf(avg + 1e-10f);
    if (0.01f*cnt > 2.0f) act += 1.f;   // cluster_size = (1-DECAY)*counts > THRESH
  }
  if (threadIdx.x < 64) vq = partials[threadIdx.x];
  rent[threadIdx.x] = ent; ract[threadIdx.x] = act; rvq[threadIdx.x] = vq;
  __syncthreads();
  for (int o = 128; o > 0; o >>= 1) {
    if (threadIdx.x < o) {
      rent[threadIdx.x] += rent[threadIdx.x+o];
      ract[threadIdx.x] += ract[threadIdx.x+o];
      rvq[threadIdx.x]  += rvq[threadIdx.x+o];
    }
    __syncthreads();
  }
  if (threadIdx.x == 0) {
    *vq_out  = 1.25f * rvq[0] / (float)(B_*CB_DIM_*T_);  // (COMMIT + CB_LOSS_W) * mean
    *ppl_out = expf(-rent[0]);
    *act_out = ract[0];
  }
}

// ---------------- out = w_out . codebook[idx] + b_out ----------------
// grid (t_tiles=8, d_splits=8, batch=8); block 256 over t
__global__ void k_outproj(const float* __restrict__ w_out, const float* __restrict__ b_out,
                          const float* __restrict__ codebook, const int* __restrict__ idx_i,
                          float* __restrict__ out) {
  __shared__ __align__(16) float wl[128*8];
  __shared__ float bl[128];
  const int d0 = blockIdx.y*128;
  for (int i = threadIdx.x; i < 128*8; i += 256) wl[i] = w_out[d0*8 + i];
  if (threadIdx.x < 128) bl[threadIdx.x] = b_out[d0 + threadIdx.x];
  __syncthreads();
  const int b = blockIdx.z;
  const int t = blockIdx.x*256 + threadIdx.x;
  const int j = idx_i[b*T_ + t];
  float cb[8];
  #pragma unroll
  for (int c = 0; c < 8; ++c) cb[c] = codebook[j*8+c];
  float* op = out + ((size_t)b*D_IN_ + d0)*T_ + t;
  for (int dl = 0; dl < 128; ++dl) {
    float s = bl[dl];
    const float* w = &wl[dl*8];
    #pragma unroll
    for (int c = 0; c < 8; ++c) s += w[c]*cb[c];
    __builtin_nontemporal_store(s, &op[(size_t)dl*T_]);  // out: write-once stream
  }
}

extern "C" void kernel_launch(void* const* d_in, const int* in_sizes, int n_in,
                              void* d_out, int out_size, void* d_ws, size_t ws_size,
                              hipStream_t stream) {
  const float* z        = (const float*)d_in[0];
  const float* codebook = (const float*)d_in[1];
  const float* v_in     = (const float*)d_in[2];
  const float* g_in     = (const float*)d_in[3];
  const float* b_in     = (const float*)d_in[4];
  const float* v_out    = (const float*)d_in[5];
  const float* g_out    = (const float*)d_in[6];
  const float* b_out    = (const float*)d_in[7];

  float* ws = (float*)d_ws;
  float* w_in   = ws + WS_WIN;
  float* w_out  = ws + WS_WOUT;
  float* cb_n   = ws + WS_CBN;
  float* ssB    = ws + WS_SSB;
  float* z_e    = ws + WS_ZE;
  float* enc    = ws + WS_ENC;
  float* ssA    = ws + WS_SSA;
  int*   idx_i  = (int*)(ws + WS_IDX);
  unsigned int* counts = (unsigned int*)(ws + WS_CNT);
  float* partials = ws + WS_PART;

  float* out_p  = (float*)d_out + OUT_OFF;
  float* idx_p  = (float*)d_out + IDX_OFF;
  float* dist_p = (float*)d_out + DIST_OFF;
  float* vq_p   = (float*)d_out + VQ_OFF;
  float* ppl_p  = (float*)d_out + PPL_OFF;
  float* act_p  = (float*)d_out + ACT_OFF;

  hipMemsetAsync(counts, 0, CB_SIZE_*sizeof(unsigned int), stream);

  k_wn_in  <<<8,    256, 0, stream>>>(v_in, g_in, w_in);
  k_wn_rows<<<36,   256, 0, stream>>>(v_out, g_out, codebook, w_out, cb_n, ssB);
  k_zproj  <<<256,  256, 0, stream>>>(z, w_in, b_in, z_e, enc, ssA);
  k_dist   <<<1024, 256, 0, stream>>>(enc, cb_n, ssA, ssB, dist_p, idx_p, idx_i, counts);
  k_vqpart <<<64,   256, 0, stream>>>(z_e, codebook, idx_i, partials);
  k_finalize<<<1,   256, 0, stream>>>(counts, partials, vq_p, ppl_p, act_p);
  k_outproj<<<dim3(8,8,8), 256, 0, stream>>>(w_out, b_out, codebook, idx_i, out_p);
}